// QuantumDenseLayer_22814866276358
// MI455X (gfx1250) — compile-verified
//
#include <hip/hip_runtime.h>

#define NQ      12
#define DIM     4096          // 2^12
#define BATCH   2048
#define DEPTH   4
#define OUT_DIM 12
#define NCOLS   32            // cols 0..11 = Re(U rows), 12..23 = Im, 24..31 = zero pad
#define KSPLIT  8
#define KCHUNK  (DIM / KSPLIT)  // 512

typedef float v2f __attribute__((ext_vector_type(2)));
typedef float v8f __attribute__((ext_vector_type(8)));

__device__ __forceinline__ float2 cmul(float2 a, float2 b) {
    return make_float2(a.x * b.x - a.y * b.y, a.x * b.y + a.y * b.x);
}
__device__ __forceinline__ float2 cadd(float2 a, float2 b) {
    return make_float2(a.x + b.x, a.y + b.y);
}

// ---------------------------------------------------------------------------
// Kernel 1: build M = first 12 rows of the circuit unitary U.
// Row j of U = (U^T e_j): apply transposed gates in reverse order to e_j.
// Block j owns one 4096-complex state in LDS (32 KB). 12 blocks x 512 threads.
// Writes B column-major: wsB[n*DIM + k], n = j -> Re, n = j+12 -> Im.
// ---------------------------------------------------------------------------
__global__ __launch_bounds__(512) void build_rows_kernel(
        const float* __restrict__ theta, float* __restrict__ wsB) {
    __shared__ float2 s[DIM];
    const int j   = blockIdx.x;   // 0..11
    const int tid = threadIdx.x;  // 0..511

    for (int k = tid; k < DIM; k += 512)
        s[k] = make_float2(k == j ? 1.0f : 0.0f, 0.0f);
    __syncthreads();

    // U = C3 R3 C2 R2 C1 R1 C0 R0  =>  U^T applied as: for l=3..0 { C_l^T ; R_l^T }
    for (int l = DEPTH - 1; l >= 0; --l) {
        const int r = (l % (NQ - 1)) + 1;

        // C_l^T : CNOT(w, (w+r)%12) for w = 11..0   (CNOT^T == CNOT)
        for (int w = NQ - 1; w >= 0; --w) {
            const int t     = (w + r) % NQ;
            const int cmask = 1 << (NQ - 1 - w);   // wire w = MSB-first bit
            const int tmask = 1 << (NQ - 1 - t);
            for (int idx = tid; idx < DIM; idx += 512) {
                if ((idx & cmask) && !(idx & tmask)) {
                    float2 a = s[idx];
                    float2 b = s[idx ^ tmask];
                    s[idx]         = b;
                    s[idx ^ tmask] = a;
                }
            }
            __syncthreads();
        }

        // R_l^T : Rot(l,w)^T for w = 11..0
        for (int w = NQ - 1; w >= 0; --w) {
            const float phi = theta[(l * NQ + w) * 3 + 0];
            const float th  = theta[(l * NQ + w) * 3 + 1];
            const float om  = theta[(l * NQ + w) * 3 + 2];
            const float ch = cosf(0.5f * th), sn = sinf(0.5f * th);
            const float A  = 0.5f * (phi + om), B = 0.5f * (phi - om);
            const float cA = cosf(A), sA = sinf(A);
            const float cB = cosf(B), sB = sinf(B);
            // Rot = [[e^{-iA}c, -e^{iB}s], [e^{-iB}s, e^{iA}c]]
            const float2 m00 = make_float2( ch * cA, -ch * sA);
            const float2 m01 = make_float2(-sn * cB, -sn * sB);
            const float2 m10 = make_float2( sn * cB, -sn * sB);
            const float2 m11 = make_float2( ch * cA,  ch * sA);

            const int sh = NQ - 1 - w;
            for (int p = tid; p < DIM / 2; p += 512) {
                const int i0 = ((p >> sh) << (sh + 1)) | (p & ((1 << sh) - 1));
                const int i1 = i0 | (1 << sh);
                const float2 s0 = s[i0], s1 = s[i1];
                // transposed apply: [[m00, m10],[m01, m11]]
                s[i0] = cadd(cmul(m00, s0), cmul(m10, s1));
                s[i1] = cadd(cmul(m01, s0), cmul(m11, s1));
            }
            __syncthreads();
        }
    }

    for (int k = tid; k < DIM; k += 512) {
        wsB[(size_t)j * DIM + k]             = s[k].x;  // Re
        wsB[(size_t)(j + OUT_DIM) * DIM + k] = s[k].y;  // Im
        if (j == 0) {
#pragma unroll
            for (int n = 2 * OUT_DIM; n < NCOLS; ++n)
                wsB[(size_t)n * DIM + k] = 0.0f;        // zero pad cols 24..31
        }
    }
}

// ---------------------------------------------------------------------------
// Kernel 2: C[b, 0..31] = x[b,:] @ B  via V_WMMA_F32_16X16X4_F32, K split 8x.
// Block = one 16-row M-tile; its 8 waves cover disjoint 512-wide K chunks.
// Per-row ||x||^2 accumulated from the same A-tile loads.
// A 16x4 layout: lane L holds A[M=L&15][K = kk + 2*(L>>4) .. +1]   (one b64)
// B  4x16 layout: lane L holds B[K = kk + 2*(L>>4) .. +1][N=L&15]  (one b64,
//                 B stored column-major so this is contiguous)
// ---------------------------------------------------------------------------
__global__ __launch_bounds__(256) void gemm_wmma_kernel(
        const float* __restrict__ x, const float* __restrict__ wsB,
        float* __restrict__ wsC, float* __restrict__ wsN) {
    const int lane  = threadIdx.x & 31;
    const int ktile = threadIdx.x >> 5;   // 0..7
    const int m0    = blockIdx.x * 16;    // batch row base
    const int kbase = ktile * KCHUNK;

    const int rowA = m0 + (lane & 15);
    const int koff = (lane >> 4) * 2;
    const float* pa  = x   + (size_t)rowA * DIM + kbase + koff;
    const float* pb0 = wsB + (size_t)(lane & 15) * DIM + kbase + koff;
    const float* pb1 = wsB + (size_t)(16 + (lane & 15)) * DIM + kbase + koff;

    v8f c0 = {0.f, 0.f, 0.f, 0.f, 0.f, 0.f, 0.f, 0.f};
    v8f c1 = {0.f, 0.f, 0.f, 0.f, 0.f, 0.f, 0.f, 0.f};
    float nrm = 0.0f;

#pragma unroll 4
    for (int kk = 0; kk < KCHUNK; kk += 4) {
        const v2f a  = *(const v2f*)(pa  + kk);
        const v2f b0 = *(const v2f*)(pb0 + kk);
        const v2f b1 = *(const v2f*)(pb1 + kk);
        nrm += a[0] * a[0] + a[1] * a[1];
        // (neg_a, A, neg_b, B, c_mod, C, reuse_a, reuse_b)
        c0 = __builtin_amdgcn_wmma_f32_16x16x4_f32(false, a, false, b0,
                                                   (short)0, c0, false, false);
        c1 = __builtin_amdgcn_wmma_f32_16x16x4_f32(false, a, false, b1,
                                                   (short)0, c1, false, false);
    }

    // lane L (<16) has K = {0,1} mod 4 partial, lane L+16 has {2,3} mod 4
    nrm += __shfl_xor(nrm, 16);
    if (lane < 16)
        wsN[(size_t)ktile * BATCH + m0 + lane] = nrm;

    // C/D layout: VGPR r, lane L -> M = r + 8*(L>>4), N = L&15
    const int n  = lane & 15;
    const int mh = (lane >> 4) * 8;
#pragma unroll
    for (int r = 0; r < 8; ++r) {
        const size_t row = (size_t)ktile * BATCH + (m0 + mh + r);
        wsC[row * NCOLS + n]      = c0[r];
        wsC[row * NCOLS + 16 + n] = c1[r];
    }
}

// ---------------------------------------------------------------------------
// Kernel 3: deterministic K-partial reduction + probs.
// out[b,j] = 2*(yr^2 + yi^2)/||x_b||^2 - 1
// ---------------------------------------------------------------------------
__global__ __launch_bounds__(256) void finalize_kernel(
        const float* __restrict__ wsC, const float* __restrict__ wsN,
        float* __restrict__ out) {
    const int t = blockIdx.x * blockDim.x + threadIdx.x;
    if (t >= BATCH * OUT_DIM) return;
    const int b = t / OUT_DIM;
    const int j = t % OUT_DIM;
    float yr = 0.f, yi = 0.f, nn = 0.f;
#pragma unroll
    for (int kt = 0; kt < KSPLIT; ++kt) {
        const size_t row = (size_t)kt * BATCH + b;
        yr += wsC[row * NCOLS + j];
        yi += wsC[row * NCOLS + OUT_DIM + j];
        nn += wsN[(size_t)kt * BATCH + b];
    }
    out[t] = 2.0f * (yr * yr + yi * yi) / nn - 1.0f;
}

extern "C" void kernel_launch(void* const* d_in, const int* in_sizes, int n_in,
                              void* d_out, int out_size, void* d_ws, size_t ws_size,
                              hipStream_t stream) {
    const float* x     = (const float*)d_in[0];   // (2048, 4096) f32
    const float* theta = (const float*)d_in[1];   // (4, 12, 3)  f32
    float* out = (float*)d_out;                   // (2048, 12)  f32

    float* wsB = (float*)d_ws;                        // 32*4096      = 512 KB
    float* wsC = wsB + (size_t)NCOLS * DIM;           // 8*2048*32    = 2 MB
    float* wsN = wsC + (size_t)KSPLIT * BATCH * NCOLS;// 8*2048       = 64 KB

    build_rows_kernel<<<OUT_DIM, 512, 0, stream>>>(theta, wsB);
    gemm_wmma_kernel<<<BATCH / 16, 256, 0, stream>>>(x, wsB, wsC, wsN);
    finalize_kernel<<<(BATCH * OUT_DIM + 255) / 256, 256, 0, stream>>>(wsC, wsN, out);
}